// InventoryOptimizer_188978561608
// MI455X (gfx1250) — compile-verified
//
#include <hip/hip_runtime.h>
#include <hip/hip_bf16.h>
#include <math.h>

#define B_  64
#define P_  4096
#define W_  16
#define D_  128
#define DW_ 64
#define W1R_STRIDE (D_ + 4)       // 132
#define W1T_STRIDE (2 * DW_ + 2)  // 130

typedef __attribute__((ext_vector_type(2))) float v2f;
typedef __attribute__((ext_vector_type(4))) float v4f;
typedef __attribute__((ext_vector_type(8))) float v8f;

// ---------------------------------------------------------------------------
// Kernel 1: emb_term[P,D] = part_emb[P,D] @ W1r[:, :D]^T + b1r
// One wave per 16x16 output tile; K-loop over 128 in steps of 4 using
// V_WMMA_F32_16X16X4_F32 (f32-native matrix pipe). Codegen verified: paired
// global_load_b64 feeding pipelined wmma.
// ---------------------------------------------------------------------------
__global__ __launch_bounds__(32)
void emb_gemm_kernel(const float* __restrict__ part_emb,
                     const float* __restrict__ W1r,
                     const float* __restrict__ b1r,
                     float* __restrict__ emb_term) {
    const int p0   = blockIdx.x * 16;
    const int d0   = blockIdx.y * 16;
    const int lane = threadIdx.x;
    const int half = lane >> 4;   // 0 or 1
    const int lm   = lane & 15;

    // A: row p0+lm of part_emb ; B: row d0+lm of W1r (B[k,n] = W1r[n,k])
    const float* arow = part_emb + (size_t)(p0 + lm) * D_;
    const float* brow = W1r      + (size_t)(d0 + lm) * W1R_STRIDE;

    v8f c = {};
#pragma unroll
    for (int kk = 0; kk < D_; kk += 4) {
        const int ko = kk + 2 * half;
        v2f a, b;
        a.x = arow[ko + 0]; a.y = arow[ko + 1];   // -> global_load_b64
        b.x = brow[ko + 0]; b.y = brow[ko + 1];
        c = __builtin_amdgcn_wmma_f32_16x16x4_f32(
                /*neg_a=*/false, a, /*neg_b=*/false, b,
                /*c_mod=*/(short)0, c, /*reuse_a=*/false, /*reuse_b=*/false);
    }

    const float bias = b1r[d0 + lm];
#pragma unroll
    for (int r = 0; r < 8; ++r) {
        // D[M,N]: M = r + 8*half, N = lm
        emb_term[(size_t)(p0 + r + 8 * half) * D_ + (d0 + lm)] = c[r] + bias;
    }
}

// ---------------------------------------------------------------------------
// Kernel 2: src_proj[16,D] = wh_emb @ W1t[:, :Dw]^T
//           tgt_proj[16,D] = wh_emb @ W1t[:, Dw:2Dw]^T
// blockIdx.x = d-tile (8), blockIdx.y = which (0=src, 1=tgt). K = 64.
// ---------------------------------------------------------------------------
__global__ __launch_bounds__(32)
void wh_gemm_kernel(const float* __restrict__ wh_emb,
                    const float* __restrict__ W1t,
                    float* __restrict__ src_proj,
                    float* __restrict__ tgt_proj) {
    const int d0    = blockIdx.x * 16;
    const int which = blockIdx.y;
    const int lane  = threadIdx.x;
    const int half  = lane >> 4;
    const int lm    = lane & 15;

    const float* arow = wh_emb + (size_t)lm * DW_;   // exactly 16 rows
    const float* brow = W1t + (size_t)(d0 + lm) * W1T_STRIDE + which * DW_;

    v8f c = {};
#pragma unroll
    for (int kk = 0; kk < DW_; kk += 4) {
        const int ko = kk + 2 * half;
        v2f a, b;
        a.x = arow[ko + 0]; a.y = arow[ko + 1];
        b.x = brow[ko + 0]; b.y = brow[ko + 1];
        c = __builtin_amdgcn_wmma_f32_16x16x4_f32(
                false, a, false, b, (short)0, c, false, false);
    }

    float* outp = which ? tgt_proj : src_proj;
#pragma unroll
    for (int r = 0; r < 8; ++r) {
        outp[(size_t)(r + 8 * half) * D_ + (d0 + lm)] = c[r];
    }
}

// ---------------------------------------------------------------------------
// Kernel P: gather the strided weight columns into contiguous float4 tables
// so the hot VALU loops issue b128 loads instead of 4-6 scalar loads per d.
//   packR[d] = { W1r[d,128], W1r[d,129], W1r[d,130], W1r[d,131] }
//   packT[d] = { W1t[d,128] (w_s), W1t[d,129] (w_d), W2t[d], b1t[d] }
// ---------------------------------------------------------------------------
__global__ __launch_bounds__(128)
void pack_kernel(const float* __restrict__ W1r,
                 const float* __restrict__ W1t,
                 const float* __restrict__ W2t,
                 const float* __restrict__ b1t,
                 v4f* __restrict__ packR,
                 v4f* __restrict__ packT) {
    const int d = threadIdx.x;   // 0..127
    v4f r;
    r.x = W1r[(size_t)d * W1R_STRIDE + D_ + 0];
    r.y = W1r[(size_t)d * W1R_STRIDE + D_ + 1];
    r.z = W1r[(size_t)d * W1R_STRIDE + D_ + 2];
    r.w = W1r[(size_t)d * W1R_STRIDE + D_ + 3];
    packR[d] = r;
    v4f t;
    t.x = W1t[(size_t)d * W1T_STRIDE + 2 * DW_ + 0];
    t.y = W1t[(size_t)d * W1T_STRIDE + 2 * DW_ + 1];
    t.z = W2t[d];
    t.w = b1t[d];
    packT[d] = t;
}

// ---------------------------------------------------------------------------
// Kernel 3: reorder_points[b,p] =
//   softplus( sum_d relu(emb_term[p,d] + s.W1r_s[d,:]) * W2r[d] + b2r )
// Vectorized by 4: one b128 emb load + four b128 packR loads + one b128 W2r
// load per 4 d -> dense FMA stream.
// ---------------------------------------------------------------------------
__global__ __launch_bounds__(256)
void reorder_kernel(const float* __restrict__ cur,
                    const float* __restrict__ dem,
                    const float* __restrict__ lt,
                    const float* __restrict__ emb_term,
                    const v4f*  __restrict__ packR,
                    const float* __restrict__ W2r,
                    const float* __restrict__ b2r,
                    float* __restrict__ out) {
    const int idx = blockIdx.x * 256 + threadIdx.x;   // < B*P
    const int p   = idx & (P_ - 1);

    const float s0 = dem[idx];
    const float s1 = lt[idx];
    const float s3 = cur[idx];
    const v4f* e4  = (const v4f*)(emb_term + (size_t)p * D_);
    const v4f* w24 = (const v4f*)W2r;

    float acc = 0.f;
#pragma unroll 4
    for (int q = 0; q < D_ / 4; ++q) {
        const v4f e  = e4[q];
        const v4f w2 = w24[q];
#pragma unroll
        for (int j = 0; j < 4; ++j) {
            const v4f w = packR[q * 4 + j];
            float h = e[j];
            h = fmaf(s0,    w.x, h);
            h = fmaf(s1,    w.y, h);
            h = fmaf(0.95f, w.z, h);
            h = fmaf(s3,    w.w, h);
            h = fmaxf(h, 0.f);
            acc = fmaf(h, w2[j], acc);
        }
    }
    const float x = acc + b2r[0];
    // numerically-stable softplus
    out[idx] = (x > 0.f) ? x + log1pf(expf(-x)) : log1pf(expf(x));
}

// ---------------------------------------------------------------------------
// Kernel 4: transfer_matrix[s,t,p]
//   h = relu(src_proj[s,:] + tgt_proj[t,:] + b1t + su*w_s + de*w_d)
//   q = min(su,de) * sigmoid(h.W2t + b2t);  masked by su>0, de>0, s!=t
// blockIdx.y = (s,t) -> d-loop operands wave-uniform; threads span p.
// Vectorized by 4 with b128 loads (b1t/W2t folded into packT).
// ---------------------------------------------------------------------------
__global__ __launch_bounds__(256)
void transfer_kernel(const float* __restrict__ wst,
                     const float* __restrict__ wdm,
                     const float* __restrict__ src_proj,
                     const float* __restrict__ tgt_proj,
                     const v4f*  __restrict__ packT,
                     const float* __restrict__ b2t,
                     float* __restrict__ out) {
    const int st = blockIdx.y;
    const int s  = st >> 4;
    const int t  = st & 15;
    const int p  = blockIdx.x * 256 + threadIdx.x;

    const float su = wst[(size_t)s * P_ + p] - wdm[(size_t)s * P_ + p];
    const float de = wdm[(size_t)t * P_ + p] - wst[(size_t)t * P_ + p];

    const v4f* sp4 = (const v4f*)(src_proj + (size_t)s * D_);
    const v4f* tp4 = (const v4f*)(tgt_proj + (size_t)t * D_);

    float acc = 0.f;
#pragma unroll 4
    for (int q = 0; q < D_ / 4; ++q) {
        const v4f sv = sp4[q];
        const v4f tv = tp4[q];
#pragma unroll
        for (int j = 0; j < 4; ++j) {
            const v4f w = packT[q * 4 + j];   // {w_s, w_d, W2t, b1t}
            float h = sv[j] + tv[j] + w.w;
            h = fmaf(su, w.x, h);
            h = fmaf(de, w.y, h);
            h = fmaxf(h, 0.f);
            acc = fmaf(h, w.z, acc);
        }
    }
    const float score = acc + b2t[0];
    const float sig   = 1.f / (1.f + expf(-score));
    const float qty   = fminf(su, de) * sig;
    const bool  mask  = (su > 0.f) && (de > 0.f) && (s != t);
    out[((size_t)(s * W_ + t)) * P_ + p] = mask ? qty : 0.f;
}

// ---------------------------------------------------------------------------
extern "C" void kernel_launch(void* const* d_in, const int* in_sizes, int n_in,
                              void* d_out, int out_size, void* d_ws, size_t ws_size,
                              hipStream_t stream) {
    const float* cur      = (const float*)d_in[0];   // (B,P)
    const float* dem      = (const float*)d_in[1];   // (B,P)
    const float* lt       = (const float*)d_in[2];   // (B,P)
    const float* wst      = (const float*)d_in[3];   // (W,P)
    const float* wdm      = (const float*)d_in[4];   // (W,P)
    const float* part_emb = (const float*)d_in[5];   // (P,D)
    const float* wh_emb   = (const float*)d_in[6];   // (W,Dw)
    const float* W1r      = (const float*)d_in[7];   // (D, D+4)
    const float* b1r      = (const float*)d_in[8];   // (D,)
    const float* W2r      = (const float*)d_in[9];   // (1,D)
    const float* b2r      = (const float*)d_in[10];  // (1,)
    const float* W1t      = (const float*)d_in[11];  // (D, 2Dw+2)
    const float* b1t      = (const float*)d_in[12];  // (D,)
    const float* W2t      = (const float*)d_in[13];  // (1,D)
    const float* b2t      = (const float*)d_in[14];  // (1,)

    // Workspace (f32): emb_term[P*D] | src_proj[W*D] | tgt_proj[W*D]
    //                  | packR[D] (v4f) | packT[D] (v4f)
    float* emb_term = (float*)d_ws;
    float* src_proj = emb_term + (size_t)P_ * D_;
    float* tgt_proj = src_proj + (size_t)W_ * D_;
    v4f*   packR    = (v4f*)(tgt_proj + (size_t)W_ * D_);
    v4f*   packT    = packR + D_;

    float* out_reorder  = (float*)d_out;                  // (B,P)
    float* out_transfer = out_reorder + (size_t)B_ * P_;  // (W,W,P)

    // 1) part_emb @ W1r_e^T + b1r  (WMMA f32)
    emb_gemm_kernel<<<dim3(P_ / 16, D_ / 16), 32, 0, stream>>>(
        part_emb, W1r, b1r, emb_term);

    // 2) wh_emb @ W1t_src^T / W1t_tgt^T  (WMMA f32)
    wh_gemm_kernel<<<dim3(D_ / 16, 2), 32, 0, stream>>>(
        wh_emb, W1t, src_proj, tgt_proj);

    // 2b) pack strided weight columns into contiguous float4 tables
    pack_kernel<<<1, 128, 0, stream>>>(W1r, W1t, W2t, b1t, packR, packT);

    // 3) reorder points (VALU, b128-vectorized)
    reorder_kernel<<<(B_ * P_) / 256, 256, 0, stream>>>(
        cur, dem, lt, emb_term, packR, W2r, b2r, out_reorder);

    // 4) transfer matrix (VALU, b128-vectorized)
    transfer_kernel<<<dim3(P_ / 256, W_ * W_), 256, 0, stream>>>(
        wst, wdm, src_proj, tgt_proj, packT, b2t, out_transfer);
}